// NemotronHTopkRouter_13580686590240
// MI455X (gfx1250) — compile-verified
//
#include <hip/hip_runtime.h>
#include <hip/hip_bf16.h>

// --- CDNA5 WMMA vector types -------------------------------------------------
typedef __attribute__((ext_vector_type(2))) float v2f;   // A/B for f32 16x16x4
typedef __attribute__((ext_vector_type(8))) float v8f;   // C/D accumulator
typedef __attribute__((ext_vector_type(4))) float f4;    // 128-bit global loads

#define TOKENS_TOTAL 16384
#define HIDDEN       4096
#define N_EXPERTS    256
#define TOK_PER_BLK  32          // 2 WMMA M-tiles per block
#define LDS_STRIDE   257         // floats; stride%64==1 -> conflict-free rows
#define NEG_INF      (-3.402823466e38f)

__device__ __forceinline__ float sigmoidf(float x) {
    return 1.0f / (1.0f + __expf(-x));
}

__global__ __launch_bounds__(256)
void router_gemm_topk(const float* __restrict__ hidden,   // [16384, 4096]
                      const float* __restrict__ weight,   // [256, 4096]
                      const float* __restrict__ bias,     // [256]
                      float* __restrict__ out)            // idx[16384*8] ++ w[16384*8]
{
    __shared__ float lds_logits[TOK_PER_BLK * LDS_STRIDE];  // 32.9 KB
    __shared__ float lds_bias[N_EXPERTS];

    const int tid  = threadIdx.x;
    const int wave = tid >> 5;        // 8 waves
    const int lane = tid & 31;
    const int hi   = lane >> 4;       // lane half selects K-slot pair
    const int lr   = lane & 15;       // row within tile
    const int T0   = blockIdx.x * TOK_PER_BLK;
    const int E0   = wave * 32;       // 32 experts per wave (2 N-tiles)

    if (tid < N_EXPERTS) lds_bias[tid] = bias[tid];

    // ---------------- fp32 WMMA GEMM: logits tile [32 tokens x 32 experts/wave]
    v8f acc00 = {}; v8f acc01 = {}; v8f acc10 = {}; v8f acc11 = {};

    // A: per-lane row pointers for the two M-tiles.
    // Lane layout for V_WMMA_F32_16X16X4_F32 (ISA 7.12.2): lanes 0-15 carry
    // K-slots {0,1}, lanes 16-31 carry K-slots {2,3}. We map physical k's
    // k0+4*hi .. k0+4*hi+3 onto the slots identically for A and B, which is a
    // legal K-permutation of the dot product.
    const float* a0 = hidden + (size_t)(T0 + lr)      * HIDDEN + 4 * hi;
    const float* a1 = hidden + (size_t)(T0 + 16 + lr) * HIDDEN + 4 * hi;
    const float* b0 = weight + (size_t)(E0 + lr)      * HIDDEN + 4 * hi;
    const float* b1 = weight + (size_t)(E0 + 16 + lr) * HIDDEN + 4 * hi;

#pragma unroll 4
    for (int k0 = 0; k0 < HIDDEN; k0 += 8) {
        f4 va0 = *(const f4*)(a0 + k0);
        f4 va1 = *(const f4*)(a1 + k0);
        f4 vb0 = *(const f4*)(b0 + k0);
        f4 vb1 = *(const f4*)(b1 + k0);

        v2f A0lo = {va0.x, va0.y}, A0hi = {va0.z, va0.w};
        v2f A1lo = {va1.x, va1.y}, A1hi = {va1.z, va1.w};
        v2f B0lo = {vb0.x, vb0.y}, B0hi = {vb0.z, vb0.w};
        v2f B1lo = {vb1.x, vb1.y}, B1hi = {vb1.z, vb1.w};

        acc00 = __builtin_amdgcn_wmma_f32_16x16x4_f32(false, A0lo, false, B0lo, (short)0, acc00, false, false);
        acc00 = __builtin_amdgcn_wmma_f32_16x16x4_f32(false, A0hi, false, B0hi, (short)0, acc00, false, false);
        acc01 = __builtin_amdgcn_wmma_f32_16x16x4_f32(false, A0lo, false, B1lo, (short)0, acc01, false, false);
        acc01 = __builtin_amdgcn_wmma_f32_16x16x4_f32(false, A0hi, false, B1hi, (short)0, acc01, false, false);
        acc10 = __builtin_amdgcn_wmma_f32_16x16x4_f32(false, A1lo, false, B0lo, (short)0, acc10, false, false);
        acc10 = __builtin_amdgcn_wmma_f32_16x16x4_f32(false, A1hi, false, B0hi, (short)0, acc10, false, false);
        acc11 = __builtin_amdgcn_wmma_f32_16x16x4_f32(false, A1lo, false, B1lo, (short)0, acc11, false, false);
        acc11 = __builtin_amdgcn_wmma_f32_16x16x4_f32(false, A1hi, false, B1hi, (short)0, acc11, false, false);
    }

    // C/D layout: VGPR v, lane L -> M = v + 8*(L>=16), N = L%16.
#pragma unroll
    for (int v = 0; v < 8; ++v) {
        int r0 = v + 8 * hi;            // token row within tile 0
        lds_logits[(r0)      * LDS_STRIDE + E0 + lr]      = acc00[v];
        lds_logits[(r0)      * LDS_STRIDE + E0 + 16 + lr] = acc01[v];
        lds_logits[(r0 + 16) * LDS_STRIDE + E0 + lr]      = acc10[v];
        lds_logits[(r0 + 16) * LDS_STRIDE + E0 + 16 + lr] = acc11[v];
    }
    __syncthreads();

    // ---------------- routing: one lane per token (wave 0 only) --------------
    if (tid < TOK_PER_BLK) {
        const float* row = lds_logits + tid * LDS_STRIDE;

        // group scores = sum of top-2 of (sigmoid(logit) + bias) per group of 32
        float gscore[8];
#pragma unroll
        for (int g = 0; g < 8; ++g) {
            float m1 = NEG_INF, m2 = NEG_INF;
            for (int i = 0; i < 32; ++i) {
                int e = (g << 5) + i;
                float sb = sigmoidf(row[e]) + lds_bias[e];
                if (sb > m1)      { m2 = m1; m1 = sb; }
                else if (sb > m2) { m2 = sb; }
            }
            gscore[g] = m1 + m2;
        }

        // top-4 groups (strict >, earliest index on tie, matching jax top_k)
        unsigned sel = 0;
        for (int r = 0; r < 4; ++r) {
            float best = NEG_INF; int bg = 0;
#pragma unroll
            for (int g = 0; g < 8; ++g)
                if (!((sel >> g) & 1u) && gscore[g] > best) { best = gscore[g]; bg = g; }
            sel |= 1u << bg;
        }

        // top-8 experts over masked (sigmoid+bias), 0.0 outside selected groups
        float tv[8]; int ti[8];
#pragma unroll
        for (int j = 0; j < 8; ++j) { tv[j] = NEG_INF; ti[j] = 0; }
        for (int e = 0; e < N_EXPERTS; ++e) {
            float sb = sigmoidf(row[e]) + lds_bias[e];
            float cv = ((sel >> (e >> 5)) & 1u) ? sb : 0.0f;
            int   ci = e;
#pragma unroll
            for (int j = 0; j < 8; ++j) {       // stable descending insertion
                if (cv > tv[j]) {
                    float fv = tv[j]; int fi = ti[j];
                    tv[j] = cv; ti[j] = ci; cv = fv; ci = fi;
                }
            }
        }

        // gather raw sigmoid scores, normalize, scale by 2.5
        float w[8], sum = 0.0f;
#pragma unroll
        for (int j = 0; j < 8; ++j) { w[j] = sigmoidf(row[ti[j]]); sum += w[j]; }
        float scale = 2.5f / (sum + 1e-20f);

        size_t gt = (size_t)T0 + tid;
#pragma unroll
        for (int j = 0; j < 8; ++j) {
            out[gt * 8 + j] = (float)ti[j];                                   // topk_indices
            out[(size_t)TOKENS_TOTAL * 8 + gt * 8 + j] = w[j] * scale;        // topk_weights
        }
    }
}

extern "C" void kernel_launch(void* const* d_in, const int* in_sizes, int n_in,
                              void* d_out, int out_size, void* d_ws, size_t ws_size,
                              hipStream_t stream) {
    const float* hidden = (const float*)d_in[0];
    const float* weight = (const float*)d_in[1];
    const float* bias   = (const float*)d_in[2];
    float* out = (float*)d_out;

    dim3 grid(TOKENS_TOTAL / TOK_PER_BLK);   // 512 blocks
    dim3 block(256);                         // 8 wave32
    hipLaunchKernelGGL(router_gemm_topk, grid, block, 0, stream,
                       hidden, weight, bias, out);
}